// gl_62792421868061
// MI455X (gfx1250) — compile-verified
//
#include <hip/hip_runtime.h>

// ---------------- problem constants (match reference) ----------------
#define GMAX   20            // GRAPHMAX
#define GVAR   40            // GRAPHVAR
#define KCON   10            // KEEPCONST
#define WCOL   (GVAR - KCON) // 30
#define XC     (GMAX + GVAR) // 60 columns of x / out
#define ITERS  2
#define WAVES  3             // waves per block (96 threads)
#define XROWS  32            // padded row count (rows 20..31 garbage-tolerant)
#define WST    48            // s_wei row stride (cols 40..47 garbage-tolerant)

typedef float v2f __attribute__((ext_vector_type(2)));
typedef float v4f __attribute__((ext_vector_type(4)));
typedef float v8f __attribute__((ext_vector_type(8)));
typedef unsigned int u32;
typedef u32 u32x4 __attribute__((ext_vector_type(4)));
typedef int  i32x4 __attribute__((ext_vector_type(4)));
typedef int  i32x8 __attribute__((ext_vector_type(8)));

#if __has_builtin(__builtin_amdgcn_tensor_load_to_lds) && \
    __has_builtin(__builtin_amdgcn_tensor_store_from_lds) && \
    __has_builtin(__builtin_amdgcn_s_wait_tensorcnt)
#define USE_TDM 1
#else
#define USE_TDM 0
#endif

// D(16x16,f32) += A(16x4,f32) * B(4x16,f32) -- CDNA5 wave32 WMMA
static __device__ __forceinline__ v8f wmma_f32(v2f a, v2f b, v8f c) {
    return __builtin_amdgcn_wmma_f32_16x16x4_f32(
        false, a, false, b, (short)0, c, false, false);
}

#if USE_TDM
static __device__ __forceinline__ u32 lds_off(const void* p) {
    // LDS byte offset within the group segment (addrspace(3) ptrs are 32-bit)
    return (u32)(unsigned long long)(const __attribute__((address_space(3))) char*)p;
}
// D# group 0: count=1 | lds_addr | global_addr[56:0] | type=2
static __device__ __forceinline__ u32x4 tdm_g0(u32 lds, const void* g) {
    unsigned long long ga = (unsigned long long)g;
    u32x4 d;
    d.x = 1u;                               // count=1, user descriptor
    d.y = lds;                              // LDS byte address
    d.z = (u32)ga;                          // global_addr[31:0]
    d.w = (u32)(ga >> 32) | (2u << 30);     // global_addr[56:32] | type=2
    return d;
}
// D# group 1: 2D tile 60x20 f32, tensor 60x20, dim0 stride 60
static __device__ __forceinline__ i32x8 tdm_g1(void) {
    i32x8 d = {};
    d[0] = (int)(2u << 16);     // data_size = 4 bytes
    d[1] = (int)(60u << 16);    // tensor_dim0[15:0] = 60
    d[2] = (int)(20u << 16);    // tensor_dim1[15:0] = 20
    d[3] = (int)(60u << 16);    // tile_dim0 = 60
    d[4] = 20;                  // tile_dim1 = 20
    d[5] = 60;                  // tensor_dim0_stride = 60
    d[6] = (int)(1200u << 16);  // tensor_dim1_stride[15:0] (unused Z)
    d[7] = 0;
    return d;
}
#endif

// stage one 20x60 f32 tile global -> LDS (tile contiguous: 1200 dwords)
static __device__ __forceinline__ void stage_in(float* lds, const float* g, int lane) {
#if USE_TDM
    i32x4 z4 = {};
    i32x8 z8 = {};
    __builtin_amdgcn_tensor_load_to_lds(tdm_g0(lds_off(lds), g), tdm_g1(),
                                        z4, z4, z8, 0);
#else
    __builtin_prefetch(g, 0, 1);
    for (int i = lane; i < (GMAX * XC) / 4; i += 32)
        ((v4f*)lds)[i] = ((const v4f*)g)[i];
#endif
}
// stage one 20x60 f32 tile LDS -> global
static __device__ __forceinline__ void stage_out(float* g, const float* lds, int lane) {
#if USE_TDM
    asm volatile("s_wait_dscnt 0x0" ::: "memory");  // val-update DS stores visible
    i32x4 z4 = {};
    i32x8 z8 = {};
    __builtin_amdgcn_tensor_store_from_lds(tdm_g0(lds_off(lds), g), tdm_g1(),
                                           z4, z4, z8, 0);
#else
    for (int i = lane; i < (GMAX * XC) / 4; i += 32)
        ((v4f*)g)[i] = ((const v4f*)lds)[i];
#endif
}

__global__ __launch_bounds__(WAVES * 32)
void graphnet_kernel(const float* __restrict__ x,
                     const float* __restrict__ nw,   // neigintact (40x30)
                     const float* __restrict__ sw,   // selfintact (40x30)
                     float* __restrict__ out,
                     int nbatch)
{
    // 3 * (2*32*60 + 32*48) * 4B = 64512 B
    __shared__ float s_x  [WAVES][2][XROWS][XC]; // x tile verbatim, double-buffered
    __shared__ float s_wei[WAVES][XROWS][WST];   // weignei scratch

    const int tid  = threadIdx.x;
    const int lane = tid & 31;
    const int wv   = tid >> 5;
    const int n    = lane & 15;
    const int half = lane >> 4;

    // ---- weight B-fragments -> registers (once per wave, ~80 VGPRs) ----
    v2f wnei[2][10], wself[2][10];
    #pragma unroll
    for (int nt = 0; nt < 2; ++nt) {
        const int   bcol = nt * 16 + n;
        const int   wc   = (bcol < WCOL) ? bcol : (WCOL - 1); // clamp address
        const float keep = (bcol < WCOL) ? 1.0f : 0.0f;       // zero pad cols
        #pragma unroll
        for (int k = 0; k < 10; ++k) {
            const int kb = 4 * k + 2 * half;
            v2f t, u;
            t.x = nw[kb * WCOL + wc] * keep;
            t.y = nw[(kb + 1) * WCOL + wc] * keep;
            u.x = sw[kb * WCOL + wc] * keep;
            u.y = sw[(kb + 1) * WCOL + wc] * keep;
            wnei[nt][k]  = t;
            wself[nt][k] = u;
        }
    }

    const long gs = (long)gridDim.x * WAVES;
    long b = (long)blockIdx.x * WAVES + wv;
    int parity = 0;

    if (b < nbatch)
        stage_in(&s_x[wv][0][0][0], x + (size_t)b * (GMAX * XC), lane);

    for (; b < nbatch; b += gs) {
        const long bn = b + gs;
        if (bn < nbatch) {
            stage_in(&s_x[wv][parity ^ 1][0][0], x + (size_t)bn * (GMAX * XC), lane);
#if USE_TDM
            __builtin_amdgcn_s_wait_tensorcnt(1);  // current load + prev store done
#endif
        } else {
#if USE_TDM
            __builtin_amdgcn_s_wait_tensorcnt(0);
#endif
        }
        asm volatile("" ::: "memory");

        float (*sx)[XC] = s_x[wv][parity];

        // ---- trace of mat (diag of cols 0..19) ----
        const int l2 = (lane < GMAX) ? lane : 0;
        float d = sx[l2][l2];
        d = (lane < GMAX) ? d : 0.0f;
        #pragma unroll
        for (int off = 16; off > 0; off >>= 1) d += __shfl_xor(d, off, 32);
        const float rtra = 1.0f / d;

        // ---- mat0 A-fragments: diag zeroed in-register (iteration-invariant) ----
        v2f amat[2][5];
        #pragma unroll
        for (int mt = 0; mt < 2; ++mt) {
            const int ar = mt * 16 + n;            // rows 20..31: garbage-tolerant
            #pragma unroll
            for (int k = 0; k < 5; ++k) {
                const int kb = 4 * k + 2 * half;
                v2f a = *(const v2f*)&sx[ar][kb];
                a.x = (ar == kb)     ? 0.0f : a.x;
                a.y = (ar == kb + 1) ? 0.0f : a.y;
                amat[mt][k] = a;
            }
        }

        for (int it = 0; it < ITERS; ++it) {
            // ---- val B-fragments: val[k][c] = sx[k][20+c] (cols >=60 garbage-ok)
            v2f bval[3][5];
            #pragma unroll
            for (int nt = 0; nt < 3; ++nt) {
                const int col = 20 + nt * 16 + n;
                #pragma unroll
                for (int k = 0; k < 5; ++k) {
                    const int kb = 4 * k + 2 * half;
                    v2f t;
                    t.x = sx[kb][col];
                    t.y = sx[kb + 1][col];
                    bval[nt][k] = t;
                }
            }
            // ========= Phase A: wei = mat0 @ val (3 interleaved chains) ========
            #pragma unroll
            for (int mt = 0; mt < 2; ++mt) {
                v8f a0 = {}, a1 = {}, a2 = {};
                #pragma unroll
                for (int k = 0; k < 5; ++k) {
                    a0 = wmma_f32(amat[mt][k], bval[0][k], a0);
                    a1 = wmma_f32(amat[mt][k], bval[1][k], a1);
                    a2 = wmma_f32(amat[mt][k], bval[2][k], a2);
                }
                const int rb = mt * 16 + 8 * half;
                #pragma unroll
                for (int v = 0; v < 8; ++v) {
                    s_wei[wv][rb + v][n]      = a0[v];
                    s_wei[wv][rb + v][16 + n] = a1[v];
                    s_wei[wv][rb + v][32 + n] = a2[v]; // cols 32..39 read, 40..47 pad
                }
            }
            asm volatile("" ::: "memory");

            // ====== Phase B: var = (wei @ neig + val @ self) * rtra ============
            v8f accs[4];
            #pragma unroll
            for (int mt = 0; mt < 2; ++mt) {
                const int ar = mt * 16 + n;
                v2f awei[10], aval[10];
                #pragma unroll
                for (int k = 0; k < 10; ++k) {
                    const int kb = 4 * k + 2 * half;
                    awei[k] = *(const v2f*)&s_wei[wv][ar][kb];
                    aval[k] = *(const v2f*)&sx[ar][20 + kb];
                }
                v8f c0 = {}, c1 = {};
                #pragma unroll
                for (int k = 0; k < 10; ++k) {
                    c0 = wmma_f32(awei[k], wnei[0][k],  c0);
                    c1 = wmma_f32(awei[k], wnei[1][k],  c1);
                    c0 = wmma_f32(aval[k], wself[0][k], c0);
                    c1 = wmma_f32(aval[k], wself[1][k], c1);
                }
                accs[mt * 2]     = c0;
                accs[mt * 2 + 1] = c1;
            }
            asm volatile("" ::: "memory");
            // in-place val update: cols 30+n always valid; cols 46+n need n<14
            #pragma unroll
            for (int mt = 0; mt < 2; ++mt) {
                const int rb = mt * 16 + 8 * half;
                #pragma unroll
                for (int v = 0; v < 8; ++v)
                    sx[rb + v][30 + n] = accs[mt * 2][v] * rtra;
            }
            if (n < 14) {   // single guarded store block (dropped cols are zero)
                #pragma unroll
                for (int mt = 0; mt < 2; ++mt) {
                    const int rb = mt * 16 + 8 * half;
                    #pragma unroll
                    for (int v = 0; v < 8; ++v)
                        sx[rb + v][46 + n] = accs[mt * 2 + 1][v] * rtra;
                }
            }
            asm volatile("" ::: "memory");
        }

        // tile now equals the output row block (mat | con | new val)
        stage_out(out + (size_t)b * (GMAX * XC), &sx[0][0], lane);
        parity ^= 1;
    }
}

extern "C" void kernel_launch(void* const* d_in, const int* in_sizes, int n_in,
                              void* d_out, int out_size, void* d_ws, size_t ws_size,
                              hipStream_t stream) {
    const float* x  = (const float*)d_in[0];
    const float* nw = (const float*)d_in[1];
    const float* sw = (const float*)d_in[2];
    float* out = (float*)d_out;
    const int nbatch = in_sizes[0] / (GMAX * XC);

    int grid = (nbatch + WAVES - 1) / WAVES;
    if (grid > 8192) grid = 8192;   // grid-stride: ~5 elements per wave
    if (grid < 1) grid = 1;

    hipLaunchKernelGGL(graphnet_kernel, dim3(grid), dim3(WAVES * 32), 0, stream,
                       x, nw, sw, out, nbatch);
}